// GivensLinearParallel_28630251995439
// MI455X (gfx1250) — compile-verified
//
#include <hip/hip_runtime.h>

#define DDIM 128
#define PMAX 64
#define ROWS_PER_BLOCK 64

typedef __attribute__((ext_vector_type(2))) float v2f;
typedef __attribute__((ext_vector_type(4))) float v4f;
typedef __attribute__((ext_vector_type(8))) float v8f;

// ---------------------------------------------------------------------------
// Kernel 1: compose the full Givens schedule into a single 128x128 matrix.
// Output is stored TRANSPOSED: Mt[col * 128 + k] == M[k][col], which is the
// layout the GEMM kernel's B-fragment loads want (contiguous in k).
//
// Parallelism: 1024 threads = 128 rows x 8 k-slices. Rotations within one
// schedule step touch disjoint column pairs, so every (row, pair) update is
// independent; barriers separate steps. LDS row stride of 129 dwords makes
// the column-indexed accesses bank-conflict-free (bank = (r + col) % 64).
// ---------------------------------------------------------------------------
__global__ __launch_bounds__(1024)
void compose_givens(const float* __restrict__ angles,
                    const float* __restrict__ mask,
                    const int*   __restrict__ II,
                    const int*   __restrict__ JJ,
                    const int*   __restrict__ AA,
                    float*       __restrict__ Mt,
                    int S)
{
    __shared__ float Msh[DDIM * (DDIM + 1)];   // stride 129 -> no bank conflicts
    __shared__ float c_sh[PMAX], s_sh[PMAX];
    __shared__ int   i_sh[PMAX], j_sh[PMAX];

    const int t  = threadIdx.x;
    const int r  = t & (DDIM - 1);   // row of M owned by this thread
    const int ks = t >> 7;           // k-slice 0..7 (8 rotations each)
    float* row = &Msh[r * (DDIM + 1)];

    // identity init: each thread covers 16 columns of its row
    for (int c = ks * 16; c < ks * 16 + 16; ++c)
        row[c] = (c == r) ? 1.0f : 0.0f;
    __syncthreads();

    for (int s = 0; s < S; ++s) {
        if (t < PMAX) {
            const int   k  = s * PMAX + t;
            const float th = angles[AA[k]] * mask[k];
            c_sh[t] = cosf(th);
            s_sh[t] = sinf(th);
            i_sh[t] = II[k];
            j_sh[t] = JJ[k];
        }
        __syncthreads();
        #pragma unroll
        for (int q = 0; q < 8; ++q) {
            const int k = ks * 8 + q;
            const int i = i_sh[k];
            if (i < DDIM) {                 // padded entries carry index D -> drop
                const int   j  = j_sh[k];
                const float c  = c_sh[k], sn = s_sh[k];
                const float mi = row[i],  mj = row[j];
                row[i] = c  * mi - sn * mj;
                row[j] = sn * mi + c  * mj;
            }
        }
        __syncthreads();
    }

    // write out transposed: Mt[col][k] = M[k][col]
    for (int c = ks * 16; c < ks * 16 + 16; ++c)
        Mt[c * DDIM + r] = row[c];
}

// ---------------------------------------------------------------------------
// Kernel 2: Y = X @ M + bias via V_WMMA_F32_16X16X4_F32.
// Block = 128 threads (4 waves) covering 64 rows; each wave owns one 16-row
// strip and sweeps all eight 16-column tiles (A fragment reused 8x).
// X tile staged in LDS (32 KB) with XOR swizzle so that the A-fragment b64
// reads (16 lanes striped over 16 rows, same column pair) hit 16 distinct
// bank pairs. B fragments stream from the 64 KB Mt, which stays resident in
// the WGP cache / L2.
// ---------------------------------------------------------------------------
__global__ __launch_bounds__(128)
void gemm_bias_wmma(const float* __restrict__ X,
                    const float* __restrict__ Mt,    // [col*128 + k]
                    const float* __restrict__ bias,
                    float*       __restrict__ Y)
{
    __shared__ float Xs[ROWS_PER_BLOCK * DDIM];      // 32 KB, column-swizzled

    const int tid  = threadIdx.x;
    const int lane = tid & 31;
    const int wave = tid >> 5;                        // 0..3
    const size_t rowBlock = (size_t)blockIdx.x * ROWS_PER_BLOCK;

    // ---- stage X tile: 64 rows x 128 cols, coalesced b128 loads ----
    {
        const v4f* __restrict__ Xv = (const v4f*)(X + rowBlock * DDIM);
        #pragma unroll
        for (int it = 0; it < 16; ++it) {
            const int lin = it * 128 + tid;          // float4 index 0..2047
            const int rr  = lin >> 5;                // row 0..63
            const int cc  = (lin & 31) << 2;         // col 0,4,...,124
            const v4f val = Xv[lin];
            *(v4f*)&Xs[rr * DDIM + (cc ^ ((rr & 15) << 2))] = val;
        }
    }
    __syncthreads();

    const int half  = lane >> 4;                     // 0 | 1
    const int lm    = lane & 15;
    const int rbase = wave * 16;
    const int swz   = lm << 2;                       // (row & 15) * 4 for our row

    const v8f vzero = {};
    v8f acc[8];
    #pragma unroll
    for (int n = 0; n < 8; ++n) acc[n] = vzero;

    const float* __restrict__ arow = &Xs[(rbase + lm) * DDIM];

    #pragma unroll 4
    for (int kk = 0; kk < 32; ++kk) {
        const int k = kk * 4 + half * 2;             // this lane's K pair base
        // A fragment: A[m][k], A[m][k+1] with m = lane%16, k = 2*(lane/16)+kk*4
        const v2f a = *(const v2f*)&arow[k ^ swz];
        #pragma unroll
        for (int n = 0; n < 8; ++n) {
            // B fragment: B[k][n], B[k+1][n] == Mt[col][k], Mt[col][k+1]
            const v2f b = *(const v2f*)&Mt[(n * 16 + lm) * DDIM + k];
            acc[n] = __builtin_amdgcn_wmma_f32_16x16x4_f32(
                /*neg_a=*/false, a, /*neg_b=*/false, b,
                /*c_mod=*/(short)0, acc[n],
                /*reuse_a=*/false, /*reuse_b=*/false);
        }
    }

    // ---- bias + store: D[m][n] at lane (n = lm), VGPR v -> m = v + 8*half ----
    #pragma unroll
    for (int n = 0; n < 8; ++n) {
        const float bv = bias[n * 16 + lm];
        float* __restrict__ yp =
            Y + (rowBlock + rbase + half * 8) * DDIM + n * 16 + lm;
        #pragma unroll
        for (int v = 0; v < 8; ++v)
            yp[(size_t)v * DDIM] = acc[n][v] + bv;
    }
}

// ---------------------------------------------------------------------------
extern "C" void kernel_launch(void* const* d_in, const int* in_sizes, int n_in,
                              void* d_out, int out_size, void* d_ws, size_t ws_size,
                              hipStream_t stream)
{
    const float* x      = (const float*)d_in[0];
    const float* angles = (const float*)d_in[1];
    const float* bias   = (const float*)d_in[2];
    const float* mask   = (const float*)d_in[3];
    const int*   Ii     = (const int*)d_in[4];
    const int*   Jj     = (const int*)d_in[5];
    const int*   Aa     = (const int*)d_in[6];

    float* y  = (float*)d_out;
    float* Mt = (float*)d_ws;                 // 128*128*4 = 64 KB scratch

    const int S     = in_sizes[4] / PMAX;     // schedule steps (127 for D=128)
    const int Brows = in_sizes[0] / DDIM;     // 8192

    compose_givens<<<1, 1024, 0, stream>>>(angles, mask, Ii, Jj, Aa, Mt, S);
    gemm_bias_wmma<<<Brows / ROWS_PER_BLOCK, 128, 0, stream>>>(x, Mt, bias, y);
}